// PyTorchBVHRouter_1108101562615
// MI455X (gfx1250) — compile-verified
//
#include <hip/hip_runtime.h>

typedef __attribute__((ext_vector_type(2))) float v2f;
typedef __attribute__((ext_vector_type(8))) float v8f;

#define TOPK 8
#define NC   16          // children per node at every level (N1=N2=N3=16)
#define WSTRIDE 2056     // padded W row stride in floats: 2056 % 64 = 8 -> rows
                         // hit different LDS banks (2048 would alias to one bank)

// One block = 256 threads = 8 waves. W (3 rows + 1 zero row) is staged in LDS
// once. Each wave computes pos for 16 rows via v_wmma_f32_16x16x4_f32 with a
// branch-free inner loop (global b64 for x, ds b64 for W), writes pos[3] to
// LDS, then 128 threads each perform the 3-level BVH routing for one row.
__global__ __launch_bounds__(256) void bvh_route_kernel(
    const float* __restrict__ x,      // [B, D]
    const float* __restrict__ W,      // [3, D]
    const float* __restrict__ l1c,    // [16, 3]
    const float* __restrict__ l2c,    // [16, 16, 3]
    const float* __restrict__ l3c,    // [256, 16, 3]
    int* __restrict__ out,            // [B, TOPK]
    int B, int D)
{
    __shared__ float wlds[4 * WSTRIDE];   // rows 0..2 = W, row 3 = zeros (~32.9 KB)
    __shared__ float pos[128][4];         // padded to 4 floats vs bank conflicts

    const int tid    = threadIdx.x;
    const int wave   = tid >> 5;
    const int lane   = tid & 31;
    const int m      = lane & 15;     // A: row-in-tile / B,C: column (N)
    const int kh     = lane >> 4;     // K half-select (0: K0..1, 1: K2..3)
    const int wgrow0 = blockIdx.x * 128;
    const int row0   = wgrow0 + wave * 16;

    // ---------- Stage W into LDS (coalesced), zero the padding row ----------
    for (int r = 0; r < 3; ++r)
        for (int i = tid; i < D; i += 256)
            wlds[r * WSTRIDE + i] = W[r * D + i];
    for (int i = tid; i < D; i += 256)
        wlds[3 * WSTRIDE + i] = 0.f;
    __syncthreads();

    // ---------- Phase 1: pos = x @ W^T via WMMA f32 16x16x4 ----------
    // Clamp row so out-of-range waves still execute WMMA with EXEC all-1s.
    int arow = row0 + m; if (arow >= B) arow = B - 1;
    const float* xptr = x + (size_t)arow * D + kh * 2;
    const float* wptr = &wlds[(m < 3 ? m : 3) * WSTRIDE + kh * 2];

    v8f c = {0.f, 0.f, 0.f, 0.f, 0.f, 0.f, 0.f, 0.f};
    #pragma unroll 16
    for (int k = 0; k < D; k += 4) {
        v2f a = *(const v2f*)(xptr + k);   // global_load_b64
        v2f b = *(const v2f*)(wptr + k);   // ds_load_b64 (zero row for m>=3)
        // C += A(16x4 f32) * B(4x16 f32)
        c = __builtin_amdgcn_wmma_f32_16x16x4_f32(
                /*neg_a=*/false, a, /*neg_b=*/false, b,
                /*c_mod=*/(short)0, c, /*reuse_a=*/false, /*reuse_b=*/false);
    }

    // C layout: lane n (0-15) holds column n; VGPR j -> row j (lanes 0-15)
    // or row 8+j (lanes 16-31). Scatter the 3 real columns to LDS.
    if (m < 3) {
        const int rbase = wave * 16 + kh * 8;
        #pragma unroll
        for (int j = 0; j < 8; ++j)
            pos[rbase + j][m] = c[j];
    }
    __syncthreads();

    // ---------- Phase 2: BVH routing, one thread per row ----------
    if (tid < 128) {
        const int row = wgrow0 + tid;
        if (row < B) {
            const float px = pos[tid][0];
            const float py = pos[tid][1];
            const float pz = pos[tid][2];

            float bd[TOPK]; int bi[TOPK];

            // Level 1: 16 centers, keep 8 nearest (sqrt monotone -> skip it).
            #pragma unroll
            for (int j = 0; j < TOPK; ++j) { bd[j] = 3.4e38f; bi[j] = 0; }
            for (int i = 0; i < NC; ++i) {
                const float dx = px - l1c[i * 3 + 0];
                const float dy = py - l1c[i * 3 + 1];
                const float dz = pz - l1c[i * 3 + 2];
                const float d  = dx * dx + dy * dy + dz * dz;
                if (d < bd[TOPK - 1]) {
                    int j = TOPK - 1;
                    while (j > 0 && d < bd[j - 1]) {
                        bd[j] = bd[j - 1]; bi[j] = bi[j - 1]; --j;
                    }
                    bd[j] = d; bi[j] = i;
                }
            }
            int l1sel[TOPK];
            #pragma unroll
            for (int j = 0; j < TOPK; ++j) l1sel[j] = bi[j];

            // Level 2: 8*16 = 128 candidates, keep 8 nearest (squared dist).
            #pragma unroll
            for (int j = 0; j < TOPK; ++j) { bd[j] = 3.4e38f; bi[j] = 0; }
            for (int a = 0; a < TOPK; ++a) {
                const float* cc = l2c + (size_t)(l1sel[a] * NC) * 3;
                for (int t = 0; t < NC; ++t) {
                    const float dx = px - cc[t * 3 + 0];
                    const float dy = py - cc[t * 3 + 1];
                    const float dz = pz - cc[t * 3 + 2];
                    const float d  = dx * dx + dy * dy + dz * dz;
                    if (d < bd[TOPK - 1]) {
                        int j = TOPK - 1;
                        while (j > 0 && d < bd[j - 1]) {
                            bd[j] = bd[j - 1]; bi[j] = bi[j - 1]; --j;
                        }
                        bd[j] = d; bi[j] = a * NC + t;
                    }
                }
            }
            int l2g[TOPK];
            #pragma unroll
            for (int j = 0; j < TOPK; ++j) {
                const int f = bi[j];
                l2g[j] = l1sel[f >> 4] * NC + (f & 15);
            }

            // Level 3: 8*16 = 128 candidates, keep 8 nearest -> expert ids.
            #pragma unroll
            for (int j = 0; j < TOPK; ++j) { bd[j] = 3.4e38f; bi[j] = 0; }
            for (int a = 0; a < TOPK; ++a) {
                const float* cc = l3c + (size_t)(l2g[a] * NC) * 3;
                for (int t = 0; t < NC; ++t) {
                    const float dx = px - cc[t * 3 + 0];
                    const float dy = py - cc[t * 3 + 1];
                    const float dz = pz - cc[t * 3 + 2];
                    const float d  = dx * dx + dy * dy + dz * dz;
                    if (d < bd[TOPK - 1]) {
                        int j = TOPK - 1;
                        while (j > 0 && d < bd[j - 1]) {
                            bd[j] = bd[j - 1]; bi[j] = bi[j - 1]; --j;
                        }
                        bd[j] = d; bi[j] = a * NC + t;
                    }
                }
            }
            #pragma unroll
            for (int j = 0; j < TOPK; ++j) {
                const int f = bi[j];
                const int expert = (l2g[f >> 4] * NC + (f & 15)) & 4095;
                out[(size_t)row * TOPK + j] = expert;
            }
        }
    }
}

extern "C" void kernel_launch(void* const* d_in, const int* in_sizes, int n_in,
                              void* d_out, int out_size, void* d_ws, size_t ws_size,
                              hipStream_t stream) {
    const float* x   = (const float*)d_in[0];
    const float* W   = (const float*)d_in[1];
    const float* l1c = (const float*)d_in[2];
    const float* l2c = (const float*)d_in[3];
    const float* l3c = (const float*)d_in[4];
    int* out = (int*)d_out;

    const int D = in_sizes[1] / 3;          // W is [3, D]
    const int B = in_sizes[0] / D;          // x is [B, D]

    const int grid = (B + 127) / 128;       // 128 rows per block
    bvh_route_kernel<<<grid, 256, 0, stream>>>(x, W, l1c, l2c, l3c, out, B, D);
}